// RecModel_11003706212749
// MI455X (gfx1250) — compile-verified
//
#include <hip/hip_runtime.h>

typedef __attribute__((ext_vector_type(2))) float v2f;
typedef __attribute__((ext_vector_type(8))) float v8f;

#define B_    256
#define T_    512
#define DIN   64
#define H_    256
#define G4H   1024
#define KTOT  320          // D_IN + H
#define NKT   (KTOT / 4)   // 80 K-tiles of 4
#define BM    16           // batch rows per workgroup (== WMMA M)
#define XHP   324          // LDS row pitch for [x | h] tile (4*m mod 64 -> conflict free)
#define NTHR  512          // 16 waves: 4 waves/SIMD for latency hiding
#define NTW   4            // N-tiles (of 16 cols) per wave

// d_out layout: rnn_outputs | logits | h | c
#define OUT_RNN    0
#define OUT_LOGITS (B_ * H_)
#define OUT_H      (OUT_LOGITS + B_ * 8)
#define OUT_C      (OUT_H + B_ * H_)

// Pack weights fragment-major: Wp[kt][j][s] = W'[4*kt+s][j], where
//   W'[k][j] = (k < 64) ? W_ih[j][k] : W_hh[j][k-64].
// A lane's whole 2-element WMMA B-fragment sits at one 8B-aligned b64, and a
// wave's 32 loads tile a contiguous 256B region -> perfect coalescing.
__global__ void wp_pack_kernel(const float* __restrict__ W_ih,
                               const float* __restrict__ W_hh,
                               float* __restrict__ Wp) {
    int idx = blockIdx.x * blockDim.x + threadIdx.x;   // 0 .. 80*1024*4-1
    if (idx >= NKT * G4H * 4) return;
    int s  = idx & 3;
    int j  = (idx >> 2) & (G4H - 1);
    int kt = idx >> 12;
    int k  = 4 * kt + s;
    Wp[idx] = (k < DIN) ? W_ih[j * DIN + k] : W_hh[j * H_ + (k - DIN)];
}

__device__ __forceinline__ float fast_sigmoid(float x) {
    return 1.0f / (1.0f + __expf(-x));
}
__device__ __forceinline__ float fast_tanh(float x) {
    float e = __expf(2.0f * x);
    return (e - 1.0f) / (e + 1.0f);
}

__global__ __launch_bounds__(NTHR)
void lstm_wmma_kernel(const float* __restrict__ x,
                      const float* __restrict__ h0,
                      const float* __restrict__ c0,
                      const float* __restrict__ bvec,
                      const float* __restrict__ W_out,
                      const float* __restrict__ b_out,
                      const float* __restrict__ Wp,
                      float* __restrict__ out) {
    __shared__ float xh[BM * XHP];        // [x_t (64) | h (256)] per row, padded pitch
    __shared__ float gates[BM * G4H];     // pre-activation gates for this batch tile

    const int tid  = threadIdx.x;         // 512 threads = 16 waves (wave32)
    const int b0   = blockIdx.x * BM;
    const int lane = tid & 31;
    const int wv   = tid >> 5;            // wave id: owns gate columns [wv*64, wv*64+64)
    const int nrow = lane & 15;           // A: matrix row m; B/D: column n (both lane&15)
    const int khalf = (lane >> 4) << 1;   // 0 for lanes 0-15, 2 for lanes 16-31

    // Pointwise-update ownership: column j, row-half mh (rows mh*8 .. mh*8+7).
    const int j_own = tid & (H_ - 1);
    const int mh    = tid >> 8;           // 0 or 1
    const int mbeg  = mh * 8;

    const float bi = bvec[j_own];
    const float bf = bvec[H_ + j_own];
    const float bg = bvec[2 * H_ + j_own];
    const float bo = bvec[3 * H_ + j_own];

    float creg[8];
#pragma unroll
    for (int r = 0; r < 8; ++r) {
        const int m = mbeg + r;
        creg[r] = c0[(b0 + m) * H_ + j_own];
        xh[m * XHP + DIN + j_own] = h0[(b0 + m) * H_ + j_own];
    }

    // x-tile staging: 16x64 floats, 2 per thread (one float2 each).
    const int xflat = tid * 2;
    const int xm = xflat >> 6;
    const int xk = xflat & 63;

    // Lane base into packed weights: float index = (kt*1024 + wv*64 + n)*4 + khalf.
    const float* __restrict__ wbase = Wp + (size_t)((wv << 6) + nrow) * 4 + khalf;

    const v8f vzero = {0.f, 0.f, 0.f, 0.f, 0.f, 0.f, 0.f, 0.f};

    for (int t = 0; t < T_; ++t) {
        // ---- stage x_t tile into LDS (coalesced float2 from global) ----
        {
            const float2* src = (const float2*)(x + ((size_t)(b0 + xm) * T_ + t) * DIN + xk);
            *(float2*)&xh[xm * XHP + xk] = *src;
        }
        __syncthreads();   // x + h(t) ready for all waves

        // ---- fused GEMM: gates = [x_t | h] @ [W_ih ; W_hh]^T  (fp32 WMMA) ----
        v8f acc[NTW];
#pragma unroll
        for (int nt = 0; nt < NTW; ++nt) acc[nt] = vzero;

        const float* __restrict__ xhA = &xh[nrow * XHP + khalf];
#pragma unroll 4
        for (int kt = 0; kt < NKT; ++kt) {
            // A fragment (16x4 f32): one b64 from LDS
            v2f a = *(const v2f*)(xhA + kt * 4);
            const float* wp = wbase + (size_t)kt * (G4H * 4);
#pragma unroll
            for (int nt = 0; nt < NTW; ++nt) {
                v2f bfrag = *(const v2f*)(wp + nt * 64);   // one b64 per fragment
                acc[nt] = __builtin_amdgcn_wmma_f32_16x16x4_f32(
                    false, a, false, bfrag, (short)0, acc[nt], false, false);
            }
        }

        // ---- scatter accumulators to LDS gate matrix (rows m, cols j) ----
        {
            const int mbase = (lane < 16) ? 0 : 8;
#pragma unroll
            for (int nt = 0; nt < NTW; ++nt) {
                const int j = (wv << 6) + nt * 16 + nrow;
#pragma unroll
                for (int r = 0; r < 8; ++r)
                    gates[(mbase + r) * G4H + j] = acc[nt][r];
            }
        }
        __syncthreads();   // all gate columns visible to all threads

        // ---- pointwise LSTM cell: thread owns column j_own, rows mbeg..mbeg+7 ----
#pragma unroll 4
        for (int r = 0; r < 8; ++r) {
            const int m = mbeg + r;
            const float* gr = &gates[m * G4H + j_own];
            float si = fast_sigmoid(gr[0]      + bi);
            float sf = fast_sigmoid(gr[H_]     + bf);
            float tg = fast_tanh   (gr[2 * H_] + bg);
            float so = fast_sigmoid(gr[3 * H_] + bo);
            float cn = sf * creg[r] + si * tg;
            creg[r] = cn;
            xh[m * XHP + DIN + j_own] = so * fast_tanh(cn);   // h(t+1) in place
        }
        // next iteration's post-x-load barrier orders these writes before the GEMM reads
    }

    __syncthreads();   // final h fully written

    // ---- outputs: rnn_outputs == h, c, and the 8-wide projection ----
#pragma unroll
    for (int r = 0; r < 8; ++r) {
        const int m = mbeg + r;
        const float hv = xh[m * XHP + DIN + j_own];
        const int gidx = (b0 + m) * H_ + j_own;
        out[OUT_RNN + gidx] = hv;
        out[OUT_H   + gidx] = hv;
        out[OUT_C   + gidx] = creg[r];
    }
    if (tid < BM * 8) {
        const int m = tid >> 3;
        const int d = tid & 7;
        float s = b_out[d];
        const float* wrow = W_out + d * H_;
        const float* hrow = &xh[m * XHP + DIN];
#pragma unroll 8
        for (int j = 0; j < H_; ++j) s += hrow[j] * wrow[j];
        out[OUT_LOGITS + (b0 + m) * 8 + d] = s;
    }
}

extern "C" void kernel_launch(void* const* d_in, const int* in_sizes, int n_in,
                              void* d_out, int out_size, void* d_ws, size_t ws_size,
                              hipStream_t stream) {
    (void)in_sizes; (void)n_in; (void)out_size; (void)ws_size;
    const float* inputs = (const float*)d_in[0];
    const float* h0     = (const float*)d_in[1];
    const float* c0     = (const float*)d_in[2];
    const float* W_ih   = (const float*)d_in[3];
    const float* W_hh   = (const float*)d_in[4];
    const float* b      = (const float*)d_in[5];
    const float* W_out  = (const float*)d_in[6];
    const float* b_out  = (const float*)d_in[7];
    float* Wp  = (float*)d_ws;                 // 80*1024*4 floats = 1.25 MB scratch
    float* out = (float*)d_out;

    hipLaunchKernelGGL(wp_pack_kernel,
                       dim3((NKT * G4H * 4 + 255) / 256), dim3(256), 0, stream,
                       W_ih, W_hh, Wp);
    hipLaunchKernelGGL(lstm_wmma_kernel,
                       dim3(B_ / BM), dim3(NTHR), 0, stream,
                       inputs, h0, c0, b, W_out, b_out, Wp, out);
}